// HardBatchMiningTripletLoss_81329500717558
// MI455X (gfx1250) — compile-verified
//
#include <hip/hip_runtime.h>

typedef __attribute__((ext_vector_type(2))) float v2f;
typedef __attribute__((ext_vector_type(8))) float v8f;

#define N_ROWS 8192
#define DIM 128
#define BLK 128
#define N_SLICES 4
#define CB_PER_SLICE (N_ROWS / BLK / N_SLICES)   // 16 column blocks per slice
#define LDS_STRIDE 132   // 132*4B rows: 16B-aligned, 132%64=4 -> conflict-free b64 frag loads
#define MARGIN 0.3f
#define FLT_BIG 3.402823466e38f

// ---------------- kernel 1: per-row squared norms ----------------
__global__ void sqnorm_kernel(const float* __restrict__ x, float* __restrict__ sq) {
    int i = blockIdx.x * blockDim.x + threadIdx.x;
    if (i < N_ROWS) {
        const float4* p = (const float4*)(x + (size_t)i * DIM);
        float s = 0.f;
#pragma unroll
        for (int k = 0; k < DIM / 4; ++k) {
            float4 v = p[k];
            s += v.x * v.x + v.y * v.y + v.z * v.z + v.w * v.w;
        }
        sq[i] = s;
    }
}

// ---------------- kernel 2: fused Gram + hard mining (per column slice) ----------------
__global__ __launch_bounds__(256)
void triplet_main_kernel(const float* __restrict__ x,
                         const int* __restrict__ tgt,
                         const float* __restrict__ sq,
                         float* __restrict__ apPart,
                         float* __restrict__ anPart) {
    __shared__ float sB[BLK * LDS_STRIDE];
    __shared__ float sSq[BLK];
    __shared__ int   sTg[BLK];

    const int tid    = threadIdx.x;
    const int lane   = tid & 31;
    const int wave   = tid >> 5;        // 8 waves, each owns 16 rows
    const int lane15 = lane & 15;
    const int khalf  = (lane >> 4) * 2; // A/B fragment K sub-offset
    const int rowBase0 = blockIdx.x * BLK;
    const int slice    = blockIdx.y;    // 4 column slices of 2048 cols each

    // ---- stage this block's row panel, extract per-lane A fragments ----
    {
        const float* gsrc = x + (size_t)rowBase0 * DIM;
#pragma unroll
        for (int i = 0; i < 16; ++i) {
            int idx = i * 256 + tid;            // float4 index into 128x128 panel
            int rr  = idx >> 5;
            int cc  = (idx & 31) * 4;
            float4 v = *(const float4*)(gsrc + rr * DIM + cc);
            *(float4*)&sB[rr * LDS_STRIDE + cc] = v;   // ds_store_b128 (row 16B-aligned)
        }
    }
    __syncthreads();

    v2f aFrag[32];
    {
        int m = wave * 16 + lane15;             // A layout: lane -> M row
#pragma unroll
        for (int ks = 0; ks < 32; ++ks)
            aFrag[ks] = *(const v2f*)&sB[m * LDS_STRIDE + ks * 4 + khalf];
    }

    // C layout: lane holds rows M = (lane>=16 ? 8 : 0) + r, r = 0..7
    const int myRowBase = rowBase0 + wave * 16 + (lane >> 4) * 8;
    float sqRow[8]; int tgRow[8];
#pragma unroll
    for (int r = 0; r < 8; ++r) {
        sqRow[r] = sq[myRowBase + r];
        tgRow[r] = tgt[myRowBase + r];
    }

    float ap[8], an[8];
#pragma unroll
    for (int r = 0; r < 8; ++r) { ap[r] = -FLT_BIG; an[r] = FLT_BIG; }

    const int cb0 = slice * CB_PER_SLICE;
    for (int cb = cb0; cb < cb0 + CB_PER_SLICE; ++cb) {
        __syncthreads();                        // protect sB before refill
        const float* gsrc = x + (size_t)cb * BLK * DIM;
#pragma unroll
        for (int i = 0; i < 16; ++i) {
            int idx = i * 256 + tid;
            int rr  = idx >> 5;
            int cc  = (idx & 31) * 4;
            float4 v = *(const float4*)(gsrc + rr * DIM + cc);
            *(float4*)&sB[rr * LDS_STRIDE + cc] = v;
        }
        if (tid < BLK) {
            sSq[tid] = sq[cb * BLK + tid];
            sTg[tid] = tgt[cb * BLK + tid];
        }
        __syncthreads();

#pragma unroll 1
        for (int nt = 0; nt < 8; ++nt) {
            const float* bbase = &sB[(nt * 16 + lane15) * LDS_STRIDE + khalf];

            v2f bFrag[32];
#pragma unroll
            for (int ks = 0; ks < 32; ++ks)
                bFrag[ks] = *(const v2f*)(bbase + ks * 4);

            // dual accumulators: two independent D->C chains on the matrix pipe
            v8f acc0 = {}, acc1 = {};
#pragma unroll
            for (int ks = 0; ks < 32; ks += 2) {
                acc0 = __builtin_amdgcn_wmma_f32_16x16x4_f32(
                        false, aFrag[ks],     false, bFrag[ks],     (short)0, acc0, false, false);
                acc1 = __builtin_amdgcn_wmma_f32_16x16x4_f32(
                        false, aFrag[ks + 1], false, bFrag[ks + 1], (short)0, acc1, false, false);
            }

            // enforce batched issue: 4 ds-reads (8 fragments) ahead of each 8-WMMA burst
#pragma unroll
            for (int g = 0; g < 4; ++g) {
                __builtin_amdgcn_sched_group_barrier(0x100, 4, 0); // 4 DS reads
                __builtin_amdgcn_sched_group_barrier(0x008, 8, 0); // 8 matrix ops
            }

            float sqC  = sSq[nt * 16 + lane15];
            int   tgC  = sTg[nt * 16 + lane15];
            int   colG = cb * BLK + nt * 16 + lane15;
#pragma unroll
            for (int r = 0; r < 8; ++r) {
                float g  = acc0[r] + acc1[r];
                float d2 = sqRow[r] + sqC - 2.0f * g;
                float d  = sqrtf(fmaxf(d2, 1e-12f));
                bool same = (tgRow[r] == tgC);
                bool self = ((myRowBase + r) == colG);
                if (same && !self) ap[r] = fmaxf(ap[r], d);
                if (!same)         an[r] = fminf(an[r], d);
            }
        }
    }

    // reduce across the 16 lanes that share each row set
#pragma unroll
    for (int r = 0; r < 8; ++r) {
#pragma unroll
        for (int m = 1; m <= 8; m <<= 1) {
            ap[r] = fmaxf(ap[r], __shfl_xor(ap[r], m, 32));
            an[r] = fminf(an[r], __shfl_xor(an[r], m, 32));
        }
    }
    if (lane15 == 0) {
#pragma unroll
        for (int r = 0; r < 8; ++r) {
            apPart[slice * N_ROWS + myRowBase + r] = ap[r];
            anPart[slice * N_ROWS + myRowBase + r] = an[r];
        }
    }
}

// ---------------- kernel 3: combine slices -> per-row hinge ----------------
__global__ void combine_kernel(const float* __restrict__ apPart,
                               const float* __restrict__ anPart,
                               float* __restrict__ lossRow) {
    int i = blockIdx.x * blockDim.x + threadIdx.x;
    if (i < N_ROWS) {
        float ap = -FLT_BIG, an = FLT_BIG;
#pragma unroll
        for (int s = 0; s < N_SLICES; ++s) {
            ap = fmaxf(ap, apPart[s * N_ROWS + i]);
            an = fminf(an, anPart[s * N_ROWS + i]);
        }
        lossRow[i] = fmaxf(ap - an + MARGIN, 0.0f);
    }
}

// ---------------- kernel 4: mean reduction ----------------
__global__ void finalize_kernel(const float* __restrict__ lossRow, float* __restrict__ out) {
    __shared__ float red[8];
    float s = 0.f;
    for (int i = threadIdx.x; i < N_ROWS; i += 256) s += lossRow[i];
#pragma unroll
    for (int m = 16; m >= 1; m >>= 1) s += __shfl_xor(s, m, 32);
    if ((threadIdx.x & 31) == 0) red[threadIdx.x >> 5] = s;
    __syncthreads();
    if (threadIdx.x == 0) {
        float t = 0.f;
#pragma unroll
        for (int w = 0; w < 8; ++w) t += red[w];
        out[0] = t / (float)N_ROWS;
    }
}

extern "C" void kernel_launch(void* const* d_in, const int* in_sizes, int n_in,
                              void* d_out, int out_size, void* d_ws, size_t ws_size,
                              hipStream_t stream) {
    const float* x   = (const float*)d_in[0];
    const int*   tgt = (const int*)d_in[1];
    float* sq      = (float*)d_ws;                       // 8192 floats
    float* apPart  = sq + N_ROWS;                        // 4*8192 floats
    float* anPart  = apPart + N_SLICES * N_ROWS;         // 4*8192 floats
    float* lossRow = anPart + N_SLICES * N_ROWS;         // 8192 floats
    float* out     = (float*)d_out;

    sqnorm_kernel<<<N_ROWS / 256, 256, 0, stream>>>(x, sq);
    dim3 grid(N_ROWS / BLK, N_SLICES);
    triplet_main_kernel<<<grid, 256, 0, stream>>>(x, tgt, sq, apPart, anPart);
    combine_kernel<<<N_ROWS / 256, 256, 0, stream>>>(apPart, anPart, lossRow);
    finalize_kernel<<<1, 256, 0, stream>>>(lossRow, out);
}